// ResNet_step_82205674045716
// MI455X (gfx1250) — compile-verified
//
#include <hip/hip_runtime.h>

typedef float v4f __attribute__((ext_vector_type(4)));
typedef float v2f __attribute__((ext_vector_type(2)));
typedef float v8f __attribute__((ext_vector_type(8)));

#define THREADS 256
#define BLOCKS  4096

// out = explicit + theta*h*relu(z_10);  resid = z_10 - out
// explicit = x + (1-theta)*h*relu(x); z_{k+1} = explicit + theta*h*relu(z_k)
// with h=1, theta=0.5 both coefficients are 0.5.
__device__ __forceinline__ float fp_elem(float x, float& resid) {
    float e = fmaf(0.5f, fmaxf(x, 0.0f), x);   // explicit part
    float z = e;
#pragma unroll
    for (int k = 0; k < 10; ++k)
        z = fmaf(0.5f, fmaxf(z, 0.0f), e);     // fixed-point iteration
    float t = 0.5f * fmaxf(z, 0.0f);
    resid = z - e - t;                         // z - explicit - theta*F(z)
    return e + t;                              // out
}

__global__ __launch_bounds__(THREADS) void resnet_step_main(
    const float* __restrict__ x, float* __restrict__ out,
    float* __restrict__ partials, long long n)
{
    const long long nv     = n >> 2;                                   // float4 count
    const long long tid    = (long long)blockIdx.x * blockDim.x + threadIdx.x;
    const long long stride = (long long)gridDim.x * blockDim.x;

    const v4f* __restrict__ xv = (const v4f*)x;
    v4f* __restrict__ ov       = (v4f*)out;

    float acc = 0.0f;
    for (long long i = tid; i < nv; i += stride) {
        __builtin_prefetch(xv + i + stride, 0, 0);        // global_prefetch_b8
        v4f v = __builtin_nontemporal_load(xv + i);       // NT: stream > L2 (192MB)
        v4f o; float r;
        o.x = fp_elem(v.x, r); acc = fmaf(r, r, acc);
        o.y = fp_elem(v.y, r); acc = fmaf(r, r, acc);
        o.z = fp_elem(v.z, r); acc = fmaf(r, r, acc);
        o.w = fp_elem(v.w, r); acc = fmaf(r, r, acc);
        __builtin_nontemporal_store(o, ov + i);
    }
    // tail (n is 4-divisible for this shape, kept for generality)
    if (tid == 0) {
        for (long long i = nv << 2; i < n; ++i) {
            float r; out[i] = fp_elem(x[i], r); acc = fmaf(r, r, acc);
        }
    }

    // ---- wave32 reduction via WMMA ones-matrix trick (full f32 precision) ----
    // A(16x4): a.x -> lanes 0..15 {M=L,K=0}, lanes 16..31 {M=L-16,K=2}; a.y=0.
    // B(4x16) = all ones  =>  D[m][n] = acc[m] + acc[m+16] for every column n.
    v2f a; a.x = acc;  a.y = 0.0f;
    v2f b; b.x = 1.0f; b.y = 1.0f;
    v8f c = {};
    c = __builtin_amdgcn_wmma_f32_16x16x4_f32(
            /*neg_a=*/false, a, /*neg_b=*/false, b,
            /*c_mod=*/(short)0, c, /*reuse_a=*/false, /*reuse_b=*/false);
    // lanes <16 hold rows 0..7 across c[0..7]; lanes >=16 hold rows 8..15
    float cs = c[0] + c[1] + c[2] + c[3] + c[4] + c[5] + c[6] + c[7];
    cs += __shfl_xor(cs, 16, 32);              // combine the two half-row sums

    __shared__ float lds[THREADS / 32];
    const int lane = threadIdx.x & 31;
    const int wave = threadIdx.x >> 5;
    if (lane == 0) lds[wave] = cs;
    __syncthreads();
    if (threadIdx.x == 0) {
        float s = 0.0f;
#pragma unroll
        for (int w = 0; w < THREADS / 32; ++w) s += lds[w];
        partials[blockIdx.x] = s;              // deterministic per-block partial
    }
}

__global__ __launch_bounds__(THREADS) void resnet_step_reduce(
    const float* __restrict__ partials, int nparts, float* __restrict__ fmin_out)
{
    __shared__ float s[THREADS];
    float a = 0.0f;
    for (int i = threadIdx.x; i < nparts; i += THREADS) a += partials[i];
    s[threadIdx.x] = a;
    __syncthreads();
#pragma unroll
    for (int off = THREADS / 2; off > 0; off >>= 1) {   // fixed-order tree
        if (threadIdx.x < off) s[threadIdx.x] += s[threadIdx.x + off];
        __syncthreads();
    }
    if (threadIdx.x == 0) *fmin_out = 0.5f * s[0];
}

extern "C" void kernel_launch(void* const* d_in, const int* in_sizes, int n_in,
                              void* d_out, int out_size, void* d_ws, size_t ws_size,
                              hipStream_t stream) {
    const float* x   = (const float*)d_in[0];
    const long long n = (long long)in_sizes[0];   // 4096*16384
    float* out       = (float*)d_out;             // out[0..n-1], fmin at out[n]
    float* partials  = (float*)d_ws;

    int blocks = BLOCKS;
    if ((size_t)blocks * sizeof(float) > ws_size) {
        blocks = (int)(ws_size / sizeof(float));
        if (blocks < 1) blocks = 1;
    }

    resnet_step_main<<<blocks, THREADS, 0, stream>>>(x, out, partials, n);
    resnet_step_reduce<<<1, THREADS, 0, stream>>>(partials, blocks, out + n);
}